// LinearFilter_13632226197602
// MI455X (gfx1250) — compile-verified
//
#include <hip/hip_runtime.h>
#include <cstdint>
#include <cstddef>

// ---------------------------------------------------------------------------
// LightGCN-style linear filter: z = A^T (A x),  A sparse COO [100000 x 50000],
// B = 64 feature columns. Scatter-SpMM. CDNA5 paths used:
//   * global_load_async_to_lds_b64 double-buffered edge staging (ASYNCcnt),
//     s_wait_asynccnt 0x3 partial waits for DMA/compute overlap
//   * hardware global_atomic_add_f32 (device scope) for the scatter
// Roofline: 0.82 GFLOP vs ~3.4 GB of gather/atomic traffic (AI ~0.25 FLOP/B)
// -> L2-bound, not matrix-core territory; x/y/z + edge stream (~89 MB) are
// resident in the 192 MB L2.
// ---------------------------------------------------------------------------

#define BDIM 64
#define N_USERS_FIXED 100000
#define EDGES_PER_WAVE 64
#define EDGES_PER_BLOCK 512   // 8 waves * 64 edges
#define SPMM_BLOCKS 1024

__global__ __launch_bounds__(256) void zero_f4(float4* __restrict__ p, long n4) {
  long i = (long)blockIdx.x * 256 + threadIdx.x;
  if (i < n4) p[i] = make_float4(0.f, 0.f, 0.f, 0.f);
}

// signal [64, nItems] (row-major) -> xT [nItems, 64]
__global__ __launch_bounds__(256) void transpose_b2i(const float* __restrict__ src,
                                                     float* __restrict__ dst,
                                                     int nItems) {
  __shared__ float tile[64][65];
  const int i0 = blockIdx.x << 6;
  const int tx = threadIdx.x & 63;
  const int ty = threadIdx.x >> 6;
  const int item = i0 + tx;
#pragma unroll
  for (int b = ty; b < 64; b += 4)
    tile[tx][b] = (item < nItems) ? src[(size_t)b * nItems + item] : 0.f;
  __syncthreads();
#pragma unroll
  for (int r = ty; r < 64; r += 4)
    if (i0 + r < nItems) dst[((size_t)(i0 + r) << 6) + tx] = tile[r][tx];
}

// zws [nItems, 64] -> out [64, nItems]
__global__ __launch_bounds__(256) void transpose_i2b(const float* __restrict__ src,
                                                     float* __restrict__ dst,
                                                     int nItems) {
  __shared__ float tile[64][65];
  const int i0 = blockIdx.x << 6;
  const int tx = threadIdx.x & 63;
  const int ty = threadIdx.x >> 6;
#pragma unroll
  for (int r = ty; r < 64; r += 4)
    tile[r][tx] = (i0 + r < nItems) ? src[((size_t)(i0 + r) << 6) + tx] : 0.f;
  __syncthreads();
  const int item = i0 + tx;
#pragma unroll
  for (int b = ty; b < 64; b += 4)
    if (item < nItems) dst[(size_t)b * nItems + item] = tile[tx][b];
}

// --- async DMA staging helpers ---------------------------------------------

// Issue 3 async b64 loads (2 edges/lane from each of dst/src/val arrays) into
// this wave's LDS slice. No wait here (caller pipelines the wait).
__device__ __forceinline__ void stage_async(const int* dstIdx, const int* srcIdx,
                                            const float* vals,
                                            int* shD, int* shS, float* shV,
                                            int ldsIdx, int eLane2) {
  const uint32_t offD = (uint32_t)(size_t)&shD[ldsIdx];
  const uint32_t offS = (uint32_t)(size_t)&shS[ldsIdx];
  const uint32_t offV = (uint32_t)(size_t)&shV[ldsIdx];
  const uint32_t vo   = (uint32_t)eLane2 * 4u;
  asm volatile(
      "global_load_async_to_lds_b64 %0, %3, %4\n\t"
      "global_load_async_to_lds_b64 %1, %3, %5\n\t"
      "global_load_async_to_lds_b64 %2, %3, %6"
      :
      : "v"(offD), "v"(offS), "v"(offV), "v"(vo),
        "s"(dstIdx), "s"(srcIdx), "s"(vals)
      : "memory");
}

__device__ __forceinline__ void wait_async_le3() {
  asm volatile("s_wait_asynccnt 0x3" ::: "memory");  // prev buffer's 3 done
}
__device__ __forceinline__ void wait_async_le0() {
  asm volatile("s_wait_asynccnt 0x0" ::: "memory");
}

// ydst[dstIdx[e]*64 + c] += vals[e] * xsrc[srcIdx[e]*64 + c]
// Wave-per-64-edge-slice, grid-stride over 512-edge chunks, ping-pong LDS
// buffers so the async-to-LDS DMA of chunk i+1 overlaps processing of chunk i.
__global__ __launch_bounds__(256) void spmm_scatter(const int* __restrict__ dstIdx,
                                                    const int* __restrict__ srcIdx,
                                                    const float* __restrict__ vals,
                                                    const float* __restrict__ xsrc,
                                                    float* __restrict__ ydst,
                                                    int nnz) {
  __shared__ int   shD[2 * EDGES_PER_BLOCK] __attribute__((aligned(16)));
  __shared__ int   shS[2 * EDGES_PER_BLOCK] __attribute__((aligned(16)));
  __shared__ float shV[2 * EDGES_PER_BLOCK] __attribute__((aligned(16)));

  const int lane  = threadIdx.x & 31;
  const int slice = (threadIdx.x >> 5) * EDGES_PER_WAVE;
  const int c2    = lane * 2;            // this lane's 2 feature columns
  const int stride = (int)gridDim.x * EDGES_PER_BLOCK;

  int  e0 = (int)blockIdx.x * EDGES_PER_BLOCK + slice;  // first edge of slice
  if (e0 >= nnz) return;                 // no barriers used: early-out is safe
  bool fullC = (e0 + EDGES_PER_WAVE) <= nnz;
  int  buf = 0;

  // stage first chunk
  if (fullC) {
    stage_async(dstIdx, srcIdx, vals, shD, shS, shV, slice + c2, e0 + c2);
  } else {
    for (int k = lane; k < EDGES_PER_WAVE; k += 32) {  // tail: guarded scalar
      const int e = e0 + k;
      shD[slice + k] = (e < nnz) ? dstIdx[e] : 0;
      shS[slice + k] = (e < nnz) ? srcIdx[e] : 0;
      shV[slice + k] = (e < nnz) ? vals[e] : 0.f;
    }
  }

  for (;;) {
    // prefetch next chunk into the other buffer
    const int  eN      = e0 + stride;
    const bool hasNext = eN < nnz;
    const bool fullN   = hasNext && (eN + EDGES_PER_WAVE) <= nnz;
    const int  nbuf    = buf ^ 1;
    const int  nbase   = nbuf * EDGES_PER_BLOCK + slice;
    if (fullN) {
      stage_async(dstIdx, srcIdx, vals, shD, shS, shV, nbase + c2, eN + c2);
      wait_async_le3();                  // async loads complete in order:
                                         // current buffer's 3 have landed
    } else {
      if (hasNext) {
        for (int k = lane; k < EDGES_PER_WAVE; k += 32) {
          const int e = eN + k;
          shD[nbase + k] = (e < nnz) ? dstIdx[e] : 0;
          shS[nbase + k] = (e < nnz) ? srcIdx[e] : 0;
          shV[nbase + k] = (e < nnz) ? vals[e] : 0.f;
        }
      }
      wait_async_le0();
    }

    // process current buffer
    const int base = buf * EDGES_PER_BLOCK + slice;
    const int kmax = fullC ? EDGES_PER_WAVE : (nnz - e0);
#pragma unroll 4
    for (int k = 0; k < kmax; ++k) {
      const int   d = shD[base + k];     // uniform LDS broadcast reads
      const int   s = shS[base + k];
      const float v = shV[base + k];
      const float2 xv = *(const float2*)(xsrc + ((size_t)s << 6) + c2);
      const float a0 = v * xv.x;
      const float a1 = v * xv.y;
      const uint32_t bo = ((uint32_t)d * 64u + (uint32_t)c2) * 4u;
      asm volatile("global_atomic_add_f32 %0, %1, %2 scope:SCOPE_DEV"
                   :: "v"(bo), "v"(a0), "s"(ydst) : "memory");
      asm volatile("global_atomic_add_f32 %0, %1, %2 scope:SCOPE_DEV"
                   :: "v"(bo + 4u), "v"(a1), "s"(ydst) : "memory");
    }

    if (!hasNext) break;
    e0 = eN;
    fullC = fullN;
    buf = nbuf;
  }
}

// ---------------------------------------------------------------------------

extern "C" void kernel_launch(void* const* d_in, const int* in_sizes, int n_in,
                              void* d_out, int out_size, void* d_ws, size_t ws_size,
                              hipStream_t stream) {
  const float* signal = (const float*)d_in[0];  // [64, nItems]
  const float* vals   = (const float*)d_in[1];  // [nnz]
  const int*   rows   = (const int*)d_in[2];    // [nnz] user index
  const int*   cols   = (const int*)d_in[3];    // [nnz] item index
  float*       out    = (float*)d_out;          // [64, nItems]

  const int nnz    = in_sizes[1];
  const int nItems = out_size / BDIM;           // 50000
  const int nUsers = N_USERS_FIXED;             // 100000 (fixed by problem)

  // workspace layout (51.2 MB): xT | y | zws
  float* xT  = (float*)d_ws;                    // nItems*64
  float* y   = xT + (size_t)nItems * BDIM;      // nUsers*64
  float* zws = y  + (size_t)nUsers * BDIM;      // nItems*64

  // zero accumulators (y and zws are contiguous) -- required every call
  const long n4 = ((long)nUsers + nItems) * BDIM / 4;
  zero_f4<<<(int)((n4 + 255) / 256), 256, 0, stream>>>((float4*)y, n4);

  const int tBlocks = (nItems + 63) / 64;
  transpose_b2i<<<tBlocks, 256, 0, stream>>>(signal, xT, nItems);

  const int nChunks = (nnz + EDGES_PER_BLOCK - 1) / EDGES_PER_BLOCK;
  const int sBlocks = (nChunks < SPMM_BLOCKS) ? nChunks : SPMM_BLOCKS;
  // pass 1: y[u] += val * xT[col]
  spmm_scatter<<<sBlocks, 256, 0, stream>>>(rows, cols, vals, xT, y, nnz);
  // pass 2: z[i] += val * y[row]
  spmm_scatter<<<sBlocks, 256, 0, stream>>>(cols, rows, vals, y, zws, nnz);

  transpose_i2b<<<tBlocks, 256, 0, stream>>>(zws, out, nItems);
}